// GCN_64055142252591
// MI455X (gfx1250) — compile-verified
//
#include <hip/hip_runtime.h>
#include <hip/hip_bf16.h>
#include <math.h>

// ---------------------------------------------------------------------------
// GCN (3-layer) for MI455X / gfx1250.
//  * GEMMs via V_WMMA_F32_16X16X4_F32 (full fp32; GEMM FLOPs ~1.8G, negligible)
//  * W staged in LDS (transposed, stride-68 words -> conflict-free b64 reads)
//  * bias + self-loop term fused into the WMMA epilogue (saves 3 full passes)
//  * edge aggregation: float4 gather + hardware fp32 global atomics
//    (all feature matrices are 25.6 MB -> resident in the 192 MB L2)
//  * layer-3 output written directly into d_out's logits slot
// ---------------------------------------------------------------------------

typedef float v2f __attribute__((ext_vector_type(2)));
typedef float v8f __attribute__((ext_vector_type(8)));

__global__ void k_zero(float* __restrict__ p, int n) {
  int i = blockIdx.x * blockDim.x + threadIdx.x;
  if (i < n) p[i] = 0.0f;
}

// deg[dst] += edge_weight  (self-loop +1 added in k_dis)
__global__ void k_deg(const int* __restrict__ dst, const float* __restrict__ ew,
                      float* __restrict__ deg, int E) {
  int e = blockIdx.x * blockDim.x + threadIdx.x;
  if (e < E) unsafeAtomicAdd(&deg[dst[e]], ew[e]);
}

// dis = rsqrt(deg + 1); nself = dis^2 (norm of the self-loop edge)
__global__ void k_dis(const float* __restrict__ deg, float* __restrict__ dis,
                      float* __restrict__ nself, int N) {
  int n = blockIdx.x * blockDim.x + threadIdx.x;
  if (n < N) {
    float d = deg[n] + 1.0f;
    float r = (d > 0.0f) ? rsqrtf(d) : 0.0f;
    dis[n]   = r;
    nself[n] = r * r;
  }
}

__global__ void k_norm(const int* __restrict__ src, const int* __restrict__ dst,
                       const float* __restrict__ ew, const float* __restrict__ dis,
                       float* __restrict__ normE, int E) {
  int e = blockIdx.x * blockDim.x + threadIdx.x;
  if (e < E) normE[e] = dis[src[e]] * ew[e] * dis[dst[e]];
}

// ---------------------------------------------------------------------------
// WMMA f32 GEMM with fused epilogue:
//   HW[n,c]  = (RELU? max(H,0):H) @ W            (for the edge gather)
//   OUT[n,c] = bias[c] + nself[n] * HW[n,c]      (init for scatter atomics)
// 256 threads = 8 waves/block; each wave owns one 16x16 tile, K stepped by 4.
// A 16x4 layout (ISA 7.12.2): lanes0-15 M=lane {v0:K0,v1:K1};
//                             lanes16-31 M=lane-16 {v0:K2,v1:K3}.
// B 4x16 mirrored; C/D: vgpr v -> M=v / v+8 by lane half, N=lane&15.
// W staged in LDS transposed, row stride 68 words: bank = (4n+k)%64, so the
// two lane halves (k vs k+2) land on disjoint banks for the b64 B reads.
// OUT must not alias H (blocks run concurrently) -> caller ping-pongs buffers.
// ---------------------------------------------------------------------------
template <int COUT, bool RELU>
__global__ void k_gemm_wmma(const float* __restrict__ H, const float* __restrict__ Wg,
                            const float* __restrict__ bias, const float* __restrict__ nself,
                            float* __restrict__ HW, float* __restrict__ OUT, int N) {
  constexpr int COLT = COUT / 16;     // col tiles per block (4 or 1)
  constexpr int ROWT = 8 / COLT;      // row tiles per block (2 or 8)
  constexpr int WSTR = 68;            // LDS row stride in words (see above)
  __shared__ float Wl[COUT * WSTR];

  // cooperative transposed W load: Wg[k*COUT+n] -> Wl[n*WSTR+k]
  for (int i = threadIdx.x; i < 64 * COUT; i += 256) {
    int k = i / COUT;
    int n = i - k * COUT;
    Wl[n * WSTR + k] = Wg[i];
  }
  __syncthreads();                    // before any divergent wave exit

  const int wave = threadIdx.x >> 5;
  const int lane = threadIdx.x & 31;
  const int rowTile = wave / COLT;
  const int colTile = wave - rowTile * COLT;
  const int m0 = (blockIdx.x * ROWT + rowTile) * 16;
  if (m0 >= N) return;                // wave-uniform: live waves keep EXEC all-1s
  const int n0 = colTile * 16;
  const int mr = lane & 15;           // A row in tile == B/D col in tile
  const int kb = (lane >> 4) << 1;    // lane-half K offset {0,2}

  v8f acc = {};
  const float* hrow = H + (size_t)(m0 + mr) * 64;
  const float* wrow = Wl + (n0 + mr) * WSTR;
#pragma unroll
  for (int k = 0; k < 64; k += 4) {
    float a0 = hrow[k + kb];
    float a1 = hrow[k + kb + 1];
    if (RELU) { a0 = fmaxf(a0, 0.0f); a1 = fmaxf(a1, 0.0f); }
    v2f a; a.x = a0; a.y = a1;
    v2f b = *(const v2f*)&wrow[k + kb];   // 8B-aligned ds_load_b64
    acc = __builtin_amdgcn_wmma_f32_16x16x4_f32(
        false, a, false, b, (short)0, acc, false, false);
  }

  const int mbase = m0 + ((lane >> 4) << 3);
  const float bcol = bias[n0 + mr];
#pragma unroll
  for (int v = 0; v < 8; ++v) {
    float hv = acc[v];
    size_t idx = (size_t)(mbase + v) * COUT + n0 + mr;
    HW[idx]  = hv;
    OUT[idx] = bcol + nself[mbase + v] * hv;
  }
}

// 16 lanes per edge, float4 gather (global_load_b128) + 4 fp32 atomics/lane
__global__ void k_scatter64(const float* __restrict__ hw, const float* __restrict__ normE,
                            const int* __restrict__ src, const int* __restrict__ dst,
                            float* __restrict__ out, int E) {
  long t = (long)blockIdx.x * blockDim.x + threadIdx.x;
  int e = (int)(t >> 4);
  if (e >= E) return;
  int f = (int)(t & 15) << 2;
  float nm = normE[e];
  float4 v = *(const float4*)(hw + (size_t)src[e] * 64 + f);
  float* op = out + (size_t)dst[e] * 64 + f;
  unsafeAtomicAdd(op + 0, nm * v.x);
  unsafeAtomicAdd(op + 1, nm * v.y);
  unsafeAtomicAdd(op + 2, nm * v.z);
  unsafeAtomicAdd(op + 3, nm * v.w);
}

// 4 lanes per edge for the 16-wide layer
__global__ void k_scatter16(const float* __restrict__ hw, const float* __restrict__ normE,
                            const int* __restrict__ src, const int* __restrict__ dst,
                            float* __restrict__ out, int E) {
  long t = (long)blockIdx.x * blockDim.x + threadIdx.x;
  int e = (int)(t >> 2);
  if (e >= E) return;
  int f = (int)(t & 3) << 2;
  float nm = normE[e];
  float4 v = *(const float4*)(hw + (size_t)src[e] * 16 + f);
  float* op = out + (size_t)dst[e] * 16 + f;
  unsafeAtomicAdd(op + 0, nm * v.x);
  unsafeAtomicAdd(op + 1, nm * v.y);
  unsafeAtomicAdd(op + 2, nm * v.z);
  unsafeAtomicAdd(op + 3, nm * v.w);
}

// logits already live in d_out slot 1; write log_softmax (slot 0) + copy (slot 2)
__global__ void k_logsoftmax16(float* __restrict__ out, int N) {
  int n = blockIdx.x * blockDim.x + threadIdx.x;
  if (n >= N) return;
  const float* lg = out + (size_t)N * 16 + (size_t)n * 16;
  float v[16];
  float mx = -3.402823466e38f;
#pragma unroll
  for (int i = 0; i < 16; ++i) { v[i] = lg[i]; mx = fmaxf(mx, v[i]); }
  float s = 0.0f;
#pragma unroll
  for (int i = 0; i < 16; ++i) s += expf(v[i] - mx);
  float lse = mx + logf(s);
  float* o0 = out + (size_t)n * 16;
  float* o2 = out + (size_t)2 * N * 16 + (size_t)n * 16;
#pragma unroll
  for (int i = 0; i < 16; ++i) {
    o0[i] = v[i] - lse;
    o2[i] = v[i];
  }
}

extern "C" void kernel_launch(void* const* d_in, const int* in_sizes, int n_in,
                              void* d_out, int out_size, void* d_ws, size_t ws_size,
                              hipStream_t stream) {
  const float* x  = (const float*)d_in[0];
  const int*   ei = (const int*)  d_in[1];  // [2, E]
  const float* ew = (const float*)d_in[2];
  const float* W1 = (const float*)d_in[3];
  const float* b1 = (const float*)d_in[4];
  const float* W2 = (const float*)d_in[5];
  const float* b2 = (const float*)d_in[6];
  const float* W3 = (const float*)d_in[7];
  const float* b3 = (const float*)d_in[8];
  float* out = (float*)d_out;

  const int N = in_sizes[0] / 64;
  const int E = in_sizes[2];
  const int* src = ei;
  const int* dst = ei + E;

  // ws (fp32): deg | dis | nself | normE | bufA(N*64) | bufB(N*64) | bufC(N*64)
  float* w     = (float*)d_ws;
  float* deg   = w;
  float* dis   = deg + N;
  float* nself = dis + N;
  float* normE = nself + N;
  float* bufA  = normE + E;                  // hw scratch (gather source)
  float* bufB  = bufA + (size_t)N * 64;      // layer-1 output
  float* bufC  = bufB + (size_t)N * 64;      // layer-2 output
  float* logits = out + (size_t)N * 16;      // d_out slot 1 (final logits)

  const int T = 256;
  const int gN = (N + T - 1) / T;
  const int gE = (E + T - 1) / T;
  const int rowTiles = (N + 15) / 16;        // 6250 for N=100000 (exact)
  const int gSc64 = (int)(((size_t)E * 16 + T - 1) / T);
  const int gSc16 = (int)(((size_t)E * 4  + T - 1) / T);

  // --- normalization ---
  k_zero<<<gN, T, 0, stream>>>(deg, N);
  k_deg <<<gE, T, 0, stream>>>(dst, ew, deg, E);
  k_dis <<<gN, T, 0, stream>>>(deg, dis, nself, N);
  k_norm<<<gE, T, 0, stream>>>(src, dst, ew, dis, normE, E);

  // --- layer 1: x -> bufB (pre-ReLU; ReLU fused into next GEMM's A load) ---
  k_gemm_wmma<64, false><<<(rowTiles + 1) / 2, 256, 0, stream>>>(
      x, W1, b1, nself, bufA, bufB, N);
  k_scatter64<<<gSc64, T, 0, stream>>>(bufA, normE, src, dst, bufB, E);

  // --- layer 2: relu(bufB) -> bufC ---
  k_gemm_wmma<64, true><<<(rowTiles + 1) / 2, 256, 0, stream>>>(
      bufB, W2, b2, nself, bufA, bufC, N);
  k_scatter64<<<gSc64, T, 0, stream>>>(bufA, normE, src, dst, bufC, E);

  // --- layer 3: relu(bufC) -> logits (directly in d_out slot 1) ---
  k_gemm_wmma<16, true><<<(rowTiles + 7) / 8, 256, 0, stream>>>(
      bufC, W3, b3, nself, bufA, logits, N);
  k_scatter16<<<gSc16, T, 0, stream>>>(bufA, normE, src, dst, logits, E);

  // --- log_softmax (slot 0) + logits copy (slot 2) ---
  k_logsoftmax16<<<gN, T, 0, stream>>>(out, N);
}